// Attention_F_32830730011398
// MI455X (gfx1250) — compile-verified
//
#include <hip/hip_runtime.h>
#include <math.h>

// ---------------- problem constants ----------------
#define Bb     4
#define DIMc   256
#define Hh     128
#define Ww     128
#define HEADSn 8
#define CHn    32            // DIM / HEADS
#define Nn     16384         // H*W
#define SZ     16777216LL    // B * DIM * N  (floats)
#define PI_F   3.14159265358979323846f

typedef float v2f __attribute__((ext_vector_type(2)));
typedef float v8f __attribute__((ext_vector_type(8)));

// ---------------- WMMA helper (fp32, 16x16x4) ----------------
__device__ __forceinline__ v8f wmma4(v2f a, v2f b, v8f c) {
  // 8 args: (neg_a, A, neg_b, B, c_mod, C, reuse_a, reuse_b)
  return __builtin_amdgcn_wmma_f32_16x16x4_f32(false, a, false, b, (short)0, c,
                                               false, false);
}

// ------------- async global -> LDS copy (CDNA5, ASYNCcnt tracked) -------------
__device__ __forceinline__ void async_g2l_b128(unsigned lds_off, const void* g) {
  asm volatile("global_load_async_to_lds_b128 %0, %1, off"
               :: "v"(lds_off), "v"((unsigned long long)(uintptr_t)g)
               : "memory");
}
__device__ __forceinline__ void wait_asynccnt0() {
  asm volatile("s_wait_asynccnt 0" ::: "memory");
}

// ---------------------------------------------------------------------------
// Generic strided / batched (complex) GEMM on the WMMA pipe.
//   C = A * B   (+ bias on real part, MODE 0 only)
// A element (m,k): Are/Aim[z*a_bs + m*lda + k]          (k contiguous)
// B element (k,n): Bre/Bim[z*b_bs + k*b_sk + n*b_sn]
// C element (m,n): Cre/Cim[z*c_bs + m*ldc + n]
// MODE 0: real*real   MODE 1: real*complex   MODE 2: complex*complex
// One wave (32 threads) per 16x16 tile; M,N % 16 == 0, K % 4 == 0.
// ---------------------------------------------------------------------------
template <int MODE>
__global__ __launch_bounds__(32) void cgemm_wmma(
    int M, int N, int K,
    const float* __restrict__ Are, const float* __restrict__ Aim,
    long lda, long a_bs,
    const float* __restrict__ Bre, const float* __restrict__ Bim,
    long b_sk, long b_sn, long b_bs,
    float* __restrict__ Cre, float* __restrict__ Cim, long ldc, long c_bs,
    const float* __restrict__ bias)
{
  const int lane = threadIdx.x;       // 0..31
  const int lo   = lane & 15;
  const int hi   = lane >> 4;
  const int n0   = blockIdx.x * 16;
  const int m0   = blockIdx.y * 16;
  const long z   = blockIdx.z;

  const float* ar = Are + z * a_bs;
  const float* br = Bre + z * b_bs;
  const float* ai = (MODE == 2) ? (Aim + z * a_bs) : nullptr;
  const float* bi = (MODE >= 1) ? (Bim + z * b_bs) : nullptr;

  // A 16x4 fragment: lane holds A[m0+lo][k0+2*hi .. +1]
  // B 4x16 fragment: lane holds B[k0+2*hi .. +1][n0+lo]
  const long aoff = (long)(m0 + lo) * lda + 2 * hi;
  const long boff = (long)(2 * hi) * b_sk + (long)(n0 + lo) * b_sn;

  v8f cr = {0.f, 0.f, 0.f, 0.f, 0.f, 0.f, 0.f, 0.f};
  v8f ci = {0.f, 0.f, 0.f, 0.f, 0.f, 0.f, 0.f, 0.f};

  for (int k = 0; k < K; k += 4) {
    v2f af, bf;
    const long ak = aoff + k;
    const long bk = boff + (long)k * b_sk;
    af.x = ar[ak];          af.y = ar[ak + 1];
    bf.x = br[bk];          bf.y = br[bk + b_sk];
    cr = wmma4(af, bf, cr);                       // Re*Re
    if (MODE >= 1) {
      v2f bif;
      bif.x = bi[bk];       bif.y = bi[bk + b_sk];
      ci = wmma4(af, bif, ci);                    // Re(A)*Im(B)
      if (MODE == 2) {
        v2f aif, naif;
        aif.x  =  ai[ak];   aif.y  =  ai[ak + 1];
        naif.x = -aif.x;    naif.y = -aif.y;      // fp32 WMMA has no A-neg; VALU negate
        cr = wmma4(naif, bif, cr);                // -Im(A)*Im(B)
        ci = wmma4(aif,  bf,  ci);                //  Im(A)*Re(B)
      }
    }
  }

  // C/D layout: VGPR i = C[m0 + i + 8*hi][n0 + lo]
  const long cb = z * c_bs + (long)(n0 + lo);
#pragma unroll
  for (int i = 0; i < 8; ++i) {
    const int m = m0 + i + 8 * hi;
    float add = 0.f;
    if (MODE == 0 && bias) add = bias[m];
    Cre[cb + (long)m * ldc] = cr[i] + add;
    if (MODE >= 1) Cim[cb + (long)m * ldc] = ci[i];
  }
}

// ---------------------------------------------------------------------------
// Batched complex 128x128x128 GEMM with B staged in LDS via async copies.
// One 8-wave workgroup per batch image; each wave owns one 16-row strip and
// sweeps the 8 N-tiles, reading B fragments from LDS (bank-conflict free).
// A element (m,k): Are/Aim[z*a_bs + m*128 + k]   (AREAL: Aim unused)
// B: contiguous row-major 128x128 per batch at  Bre/Bim[z*b_bs]
// C element (m,n): Cre/Cim[z*c_bs + m*128 + n]
// Dynamic LDS: 2 * 16384 floats = 128 KB.
// ---------------------------------------------------------------------------
template <bool AREAL>
__global__ __launch_bounds__(256) void cgemm128_lds(
    const float* __restrict__ Are, const float* __restrict__ Aim, long a_bs,
    const float* __restrict__ Bre, const float* __restrict__ Bim, long b_bs,
    float* __restrict__ Cre, float* __restrict__ Cim, long c_bs)
{
  extern __shared__ float smem[];            // [0,16384) = B.re, [16384,32768) = B.im
  float* sBre = smem;
  float* sBim = smem + 16384;

  const long z  = blockIdx.x;
  const int  t  = threadIdx.x;

  // ---- stage B (complex 128x128 = 128 KB) into LDS, ASYNCcnt tracked ----
  {
    const float* gbr = Bre + z * b_bs;
    const float* gbi = Bim + z * b_bs;
#pragma unroll
    for (int i = 0; i < 16; ++i) {
      const int off = (i * 256 + t) * 4;     // float index, 16-byte aligned
      async_g2l_b128((unsigned)(uintptr_t)(&sBre[off]), gbr + off);
      async_g2l_b128((unsigned)(uintptr_t)(&sBim[off]), gbi + off);
    }
  }
  wait_asynccnt0();
  __syncthreads();

  const int wv   = t >> 5;                   // wave id 0..7 -> M-tile
  const int lane = t & 31;
  const int lo   = lane & 15;
  const int hi   = lane >> 4;
  const int m0   = wv * 16;

  const float* ar = Are + z * a_bs;
  const float* ai = AREAL ? nullptr : (Aim + z * a_bs);
  const long aoff = (long)(m0 + lo) * 128 + 2 * hi;

  for (int nt = 0; nt < 8; ++nt) {
    const int n0 = nt * 16;
    v8f cr = {0.f, 0.f, 0.f, 0.f, 0.f, 0.f, 0.f, 0.f};
    v8f ci = {0.f, 0.f, 0.f, 0.f, 0.f, 0.f, 0.f, 0.f};
    for (int k = 0; k < 128; k += 4) {
      v2f af, bf, bif;
      af.x = ar[aoff + k];
      af.y = ar[aoff + k + 1];
      const int brow = (k + 2 * hi) * 128 + n0 + lo;
      bf.x  = sBre[brow];   bf.y  = sBre[brow + 128];
      bif.x = sBim[brow];   bif.y = sBim[brow + 128];
      cr = wmma4(af, bf, cr);
      ci = wmma4(af, bif, ci);
      if (!AREAL) {
        v2f aif, naif;
        aif.x  =  ai[aoff + k];
        aif.y  =  ai[aoff + k + 1];
        naif.x = -aif.x;  naif.y = -aif.y;
        cr = wmma4(naif, bif, cr);
        ci = wmma4(aif,  bf,  ci);
      }
    }
    const long cb = z * c_bs + (long)(n0 + lo);
#pragma unroll
    for (int i = 0; i < 8; ++i) {
      const long mrow = (long)(m0 + i + 8 * hi) * 128;
      Cre[cb + mrow] = cr[i];
      Cim[cb + mrow] = ci[i];
    }
  }
}

// ---------------- DFT / twiddle table builder ----------------
__global__ void init_tables(float* Fre, float* Fim, float* iFre, float* iFim,
                            float* twr, float* twi, float* i32r, float* i32i)
{
  const int j = blockIdx.x * 256 + threadIdx.x;
  if (j < 16384) {
    const int r = j >> 7, c = j & 127;
    const float a = -2.f * PI_F * (float)((r * c) & 127) * (1.f / 128.f);
    Fre[j]  = cosf(a);  Fim[j]  = sinf(a);     // forward DFT_128
    iFre[j] = cosf(a);  iFim[j] = -sinf(a);    // inverse DFT_128 (unscaled)
    // four-step twiddle for N=16384: w^{+k1*n2} * (1/16384)
    const float at = 2.f * PI_F * (float)((r * c) & 16383) * (1.f / 16384.f);
    twr[j] = cosf(at) * (1.f / 16384.f);
    twi[j] = sinf(at) * (1.f / 16384.f);
  }
  if (j < 1024) {
    const int r = j >> 5, c = j & 31;
    const float a = 2.f * PI_F * (float)((r * c) & 31) * (1.f / 32.f);
    i32r[j] = cosf(a) * (1.f / 32.f);          // inverse DFT_32 incl. 1/32
    i32i[j] = sinf(a) * (1.f / 32.f);
  }
}

// ---------------- complex L2 normalize along last dim (len N) ----------------
__global__ void l2norm_rows(float* __restrict__ re, float* __restrict__ im)
{
  __shared__ float red[256];
  const long base = (long)blockIdx.x * Nn;
  float s = 0.f;
  for (int i = threadIdx.x; i < Nn; i += 256) {
    const float a = re[base + i], b = im[base + i];
    s += a * a + b * b;
  }
  red[threadIdx.x] = s;
  __syncthreads();
  for (int off = 128; off > 0; off >>= 1) {
    if (threadIdx.x < off) red[threadIdx.x] += red[threadIdx.x + off];
    __syncthreads();
  }
  const float inv = 1.f / fmaxf(sqrtf(red[0]), 1e-12f);
  for (int i = threadIdx.x; i < Nn; i += 256) {
    re[base + i] *= inv;
    im[base + i] *= inv;
  }
}

// ------- temperature scale + softmax over last dim (32) of re & im planes -------
__global__ void scale_softmax(float* __restrict__ are, float* __restrict__ aim,
                              const float* __restrict__ temp)
{
  const int row  = blockIdx.x * 8 + (threadIdx.x >> 5);   // 1024 rows total
  const int lane = threadIdx.x & 31;
  if (row >= Bb * HEADSn * CHn) return;
  const float t = temp[(row / CHn) % HEADSn];
  const long idx = (long)row * CHn + lane;
#pragma unroll
  for (int plane = 0; plane < 2; ++plane) {
    float* p = plane ? aim : are;
    float v = p[idx] * t;
    float m = v;
    for (int off = 16; off > 0; off >>= 1) m = fmaxf(m, __shfl_xor(m, off, 32));
    const float e = expf(v - m);
    float s = e;
    for (int off = 16; off > 0; off >>= 1) s += __shfl_xor(s, off, 32);
    p[idx] = e / s;
  }
}

// ---------------- four-step twiddle multiply (in place, folds 1/16384) --------
__global__ void twiddle_mul(float* __restrict__ re, float* __restrict__ im,
                            const float* __restrict__ twr,
                            const float* __restrict__ twi)
{
  const long e = (long)blockIdx.x * 256 + threadIdx.x;    // over SZ
  const int  p = (int)(e & (Nn - 1));                     // p = k1*128 + n2
  const float tr = twr[p], ti = twi[p];
  const float a = re[e], b = im[e];
  re[e] = a * tr - b * ti;
  im[e] = a * ti + b * tr;
}

// ------ |z| + four-step output permutation + [b,head,ch,n] -> [b,c,h,w] -------
__global__ void abs_transpose(const float* __restrict__ tre,
                              const float* __restrict__ tim,
                              float* __restrict__ out)
{
  const long e = (long)blockIdx.x * 256 + threadIdx.x;    // over SZ, [b,c,h,w]
  const long chan = e >> 14;
  const int  p = (int)(e & 16383);
  const int  h = p >> 7, w = p & 127;
  // n-index m = h*128+w lives at four-step matrix slot (k1=m%128, k2=m/128)
  const long src = (chan << 14) + (long)(w * 128 + h);
  const float a = tre[src], b = tim[src];
  out[e] = sqrtf(a * a + b * b);
}

// ---------------------------------------------------------------------------
extern "C" void kernel_launch(void* const* d_in, const int* in_sizes, int n_in,
                              void* d_out, int out_size, void* d_ws, size_t ws_size,
                              hipStream_t stream)
{
  const float* x    = (const float*)d_in[0];
  const float* w1   = (const float*)d_in[1];
  const float* b1   = (const float*)d_in[2];
  const float* w2   = (const float*)d_in[3];
  const float* b2   = (const float*)d_in[4];
  const float* w3   = (const float*)d_in[5];
  const float* b3   = (const float*)d_in[6];
  const float* wo   = (const float*)d_in[7];
  const float* bo   = (const float*)d_in[8];
  const float* temp = (const float*)d_in[9];
  float* out = (float*)d_out;

  float* ws = (float*)d_ws;
  // small tables / attention buffers
  float* Fre  = ws;            float* Fim  = ws + 16384;
  float* iFre = ws + 32768;    float* iFim = ws + 49152;
  float* twr  = ws + 65536;    float* twi  = ws + 81920;
  float* i32r = ws + 98304;    float* i32i = ws + 99328;
  float* a_re = ws + 102400;   float* a_im = ws + 135168;   // 4*8*32*32 each
  // big buffers (SZ floats each)
  float* big  = ws + 262144;
  float* r0   = big;                 // real staging  [b,c,h,w]
  float* t_re = big + 1 * SZ;        // complex ping
  float* t_im = big + 2 * SZ;
  float* q_re = big + 3 * SZ;  float* q_im = big + 4 * SZ;
  float* k_re = big + 5 * SZ;  float* k_im = big + 6 * SZ;
  float* v_re = big + 7 * SZ;  float* v_im = big + 8 * SZ;
  float* o_re = big + 9 * SZ;  float* o_im = big + 10 * SZ;

  const long DN = (long)DIMc * Nn;   // per-batch plane stride
  const long CN = (long)CHn * Nn;    // per-(b,head) stride
  const unsigned LDSB = 2u * 16384u * 4u;   // 128 KB dynamic LDS

  init_tables<<<64, 256, 0, stream>>>(Fre, Fim, iFre, iFim, twr, twi, i32r, i32i);

  const float* wv[3]  = {w1, w2, w3};
  const float* bv[3]  = {b1, b2, b3};
  float*       xr[3]  = {q_re, k_re, v_re};
  float*       xi[3]  = {q_im, k_im, v_im};

  for (int t = 0; t < 3; ++t) {
    // 1x1 conv:  r0[b,o,n] = W[o,c] * x[b,c,n] + bias[o]
    cgemm_wmma<0><<<dim3(Nn / 16, DIMc / 16, Bb), 32, 0, stream>>>(
        DIMc, Nn, DIMc,
        wv[t], nullptr, (long)DIMc, 0L,
        x, nullptr, (long)Nn, 1L, DN,
        r0, nullptr, (long)Nn, DN, bv[t]);
    // FFT along w:  T = X_rows * F, batched per 128-row group (A real, B=F in LDS)
    cgemm128_lds<true><<<dim3(Bb * DIMc), 256, LDSB, stream>>>(
        r0, nullptr, (long)Nn,
        Fre, Fim, 0L,
        t_re, t_im, (long)Nn);
    // FFT along h:  Y = F * T, batched per image (B = image in LDS)
    cgemm128_lds<false><<<dim3(Bb * DIMc), 256, LDSB, stream>>>(
        Fre, Fim, 0L,
        t_re, t_im, (long)Nn,
        xr[t], xi[t], (long)Nn);
  }

  // complex L2 normalize q and k rows
  l2norm_rows<<<Bb * HEADSn * CHn, 256, 0, stream>>>(q_re, q_im);
  l2norm_rows<<<Bb * HEADSn * CHn, 256, 0, stream>>>(k_re, k_im);

  // attn = Q * K^T  (complex, K-dim = 16384), batched over (b,head)
  cgemm_wmma<2><<<dim3(2, 2, Bb * HEADSn), 32, 0, stream>>>(
      CHn, CHn, Nn,
      q_re, q_im, (long)Nn, CN,
      k_re, k_im, 1L, (long)Nn, CN,         // B[k,n] = K[n,k]  (transpose via strides)
      a_re, a_im, (long)CHn, (long)CHn * CHn, nullptr);

  scale_softmax<<<(Bb * HEADSn * CHn) / 8, 256, 0, stream>>>(a_re, a_im, temp);

  // out = attn * V  (complex, 32x16384, K=32), batched over (b,head)
  cgemm_wmma<2><<<dim3(Nn / 16, CHn / 16, Bb * HEADSn), 32, 0, stream>>>(
      CHn, Nn, CHn,
      a_re, a_im, (long)CHn, (long)CHn * CHn,
      v_re, v_im, (long)Nn, 1L, CN,
      o_re, o_im, (long)Nn, CN, nullptr);

  // ifft along c (len 32, incl. 1/32):  t = iF32 * out
  cgemm_wmma<2><<<dim3(Nn / 16, CHn / 16, Bb * HEADSn), 32, 0, stream>>>(
      CHn, Nn, CHn,
      i32r, i32i, (long)CHn, 0L,
      o_re, o_im, (long)Nn, 1L, CN,
      t_re, t_im, (long)Nn, CN, nullptr);

  // ifft along n (len 16384) via four-step, per row viewed as 128x128:
  // stage 1: G = iF128 * Xm   (B = row-matrix in LDS)
  cgemm128_lds<false><<<dim3(Bb * HEADSn * CHn), 256, LDSB, stream>>>(
      iFre, iFim, 0L,
      t_re, t_im, (long)Nn,
      o_re, o_im, (long)Nn);
  // stage 2: twiddle (folds 1/16384)
  twiddle_mul<<<(unsigned)(SZ / 256), 256, 0, stream>>>(o_re, o_im, twr, twi);
  // stage 3: H = G * iF128   (B = iF128 in LDS, A streamed per image)
  cgemm128_lds<false><<<dim3(Bb * HEADSn * CHn), 256, LDSB, stream>>>(
      o_re, o_im, (long)Nn,
      iFre, iFim, 0L,
      t_re, t_im, (long)Nn);

  // |z| + output permutation -> r0 in [b,c,h,w]
  abs_transpose<<<(unsigned)(SZ / 256), 256, 0, stream>>>(t_re, t_im, r0);

  // final 1x1 conv -> d_out
  cgemm_wmma<0><<<dim3(Nn / 16, DIMc / 16, Bb), 32, 0, stream>>>(
      DIMc, Nn, DIMc,
      wo, nullptr, (long)DIMc, 0L,
      r0, nullptr, (long)Nn, 1L, DN,
      out, nullptr, (long)Nn, DN, bo);
}